// EKVConv2d_27144193310998
// MI455X (gfx1250) — compile-verified
//
#include <hip/hip_runtime.h>
#include <math.h>

// EKVConv2d: out[b,o,l] = ALPHA * sum_k [ sp((p-th)*inv)^2 - sp((p-th)*inv - VD*inv)^2 ]
// B=8, C=16, H=W=32, O=64, 3x3 pad=1. Transcendental-bound: 75.5M softplus-pairs
// = ~302M v_exp_f32/v_log_f32. Strategy: TDM-stage x tile + theta slice into LDS,
// compute in log2 domain (native v_exp/v_log are exp2/log2) with all scale factors
// folded into the staged theta and the final alpha.

#define B_N    8
#define C_IN   16
#define HW     32
#define O_TOT  64
#define OGRP   8           // output channels per block -> 256 blocks total
#define KK     144         // C_IN * 9

typedef unsigned int u32x4 __attribute__((ext_vector_type(4)));
typedef int          i32x8 __attribute__((ext_vector_type(8)));
typedef int          i32x4 __attribute__((ext_vector_type(4)));

#if defined(__AMDGCN__) && __has_builtin(__builtin_amdgcn_tensor_load_to_lds)
#define USE_TDM 1
#else
#define USE_TDM 0
#endif

// ---- constants (log2-domain formulation) ----
// INV   = 1/(N_SLOPE*VT) = 1/0.039
// SC    = INV * log2(e)            : pre-scale for patches and theta
// DLT2  = VD * INV * log2(e)       : shift between the two softplus args
// CLIP2 = 30 * log2(e)             : reference clips at +-30 (nat-log domain)
// OUTSC = ALPHA * ln(2)^2          : converts log2^2 back to softplus^2
#define SC_F    36.99218055f
#define DLT2_F  3.699218055f
#define CLIP2_F 43.28085123f
#define OUTSC_F 2.7025482e-4f

__device__ __forceinline__ float clamp2(float x) {
#if defined(__AMDGCN__) && __has_builtin(__builtin_amdgcn_fmed3f)
  return __builtin_amdgcn_fmed3f(x, -CLIP2_F, CLIP2_F);   // v_med3_f32: 1 op
#else
  return fminf(fmaxf(x, -CLIP2_F), CLIP2_F);
#endif
}

__device__ __forceinline__ float exp2_raw(float x) {
#if defined(__AMDGCN__) && __has_builtin(__builtin_amdgcn_exp2f)
  return __builtin_amdgcn_exp2f(x);       // raw v_exp_f32 (input pre-clamped)
#else
  return exp2f(x);
#endif
}

__device__ __forceinline__ float log2_raw(float x) {
#if defined(__AMDGCN__) && __has_builtin(__builtin_amdgcn_logf)
  return __builtin_amdgcn_logf(x);        // raw v_log_f32 (input >= 1 here)
#else
  return log2f(x);
#endif
}

#if USE_TDM
// Build a D# for a <=3D tile (data_size = 4 bytes) and issue TENSOR_LOAD_TO_LDS.
// Walk order: X (contiguous, tile_d0) fastest, then Y (stride s0), then Z (stride s1).
// LDS writes are fully compacted: [Z][Y][X].
__device__ __forceinline__ void tdm_load_3d(
    unsigned lds_byte, unsigned long long gaddr,
    unsigned td0, unsigned td1, unsigned td2,       // tensor dims (elements)
    unsigned tl0, unsigned tl1, unsigned tl2,       // tile dims  (elements)
    unsigned long long s0, unsigned long long s1)   // dim0/dim1 strides (elements)
{
  u32x4 g0;
  g0.x = 1u;                                               // count=1, no gather
  g0.y = lds_byte;                                         // lds_addr [63:32]
  g0.z = (unsigned)(gaddr & 0xffffffffull);                // global_addr [95:64]
  g0.w = (unsigned)((gaddr >> 32) & 0x01ffffffull)         // global_addr [120:96]
       | (2u << 30);                                       // type=2 ("image")
  i32x8 g1;
  g1[0] = (int)(2u << 16);                                 // wg_mask=0, data_size=4B
  g1[1] = (int)((td0 & 0xffffu) << 16);                    // barrier_addr=0 | td0.lo
  g1[2] = (int)(((td0 >> 16) & 0xffffu) | ((td1 & 0xffffu) << 16));
  g1[3] = (int)(((td1 >> 16) & 0xffffu) | ((tl0 & 0xffffu) << 16));
  g1[4] = (int)((tl1 & 0xffffu) | ((tl2 & 0xffffu) << 16));
  g1[5] = (int)(unsigned)(s0 & 0xffffffffull);             // tensor_dim0_stride lo
  g1[6] = (int)(((unsigned)((s0 >> 32) & 0xffffull))
              | (((unsigned)(s1 & 0xffffull)) << 16));     // s0.hi | s1.lo
  g1[7] = (int)(unsigned)((s1 >> 16) & 0xffffffffull);     // s1 [47:16]
  i32x4 g2;
  g2[0] = (int)td2;                                        // tensor_dim2
  g2[1] = 0; g2[2] = 0; g2[3] = 0;                         // no dim3 / iterate
  i32x4 g3 = {0, 0, 0, 0};
#if __clang_major__ >= 23
  i32x8 gx = {0, 0, 0, 0, 0, 0, 0, 0};
  __builtin_amdgcn_tensor_load_to_lds(g0, g1, g2, g3, gx, 0);
#else
  __builtin_amdgcn_tensor_load_to_lds(g0, g1, g2, g3, 0);
#endif
}
#endif

__global__ __launch_bounds__(256) void ekv_conv2d_kernel(
    const float* __restrict__ x,      // (8,16,32,32)
    const float* __restrict__ theta,  // (64,144)
    float* __restrict__ out)          // (8,64,32,32)
{
  __shared__ float thS[OGRP * KK];        //  4608 B: theta slice (pre-scaled by SC)
  __shared__ float xs[C_IN * 10 * HW];    // 20480 B: x rows [r0-1, r0+8] clipped

  const int tid = threadIdx.x;
  const int blk = blockIdx.x;             // 256 blocks
  const int b   = blk >> 5;               // 0..7
  const int og  = (blk >> 2) & 7;         // 0..7  (8 channels each)
  const int rq  = blk & 3;                // 0..3  (8 rows each)
  const int r0  = rq * 8;
  const int rowBase = (r0 == 0) ? 0 : (r0 - 1);
  const int rowEnd  = (r0 + 8 >= HW) ? (HW - 1) : (r0 + 8);
  const int nrows   = rowEnd - rowBase + 1;          // 9 or 10

#if USE_TDM
  {
    // Each of the 8 waves DMAs its own slice: 1 theta row + 2 x channels.
    const int w = __builtin_amdgcn_readfirstlane(tid >> 5);   // wave id 0..7
    const unsigned thL = (unsigned)(unsigned long long)(const void*)thS
                       + (unsigned)(w * KK * 4);
    tdm_load_3d(thL,
                (unsigned long long)(const void*)(theta + (og * OGRP + w) * KK),
                /*td*/ KK, 1, 1, /*tl*/ KK, 1, 1,
                /*s0*/ KK, /*s1*/ 0);
    const unsigned xL = (unsigned)(unsigned long long)(const void*)xs
                      + (unsigned)(w * 2 * nrows * HW * 4);
    tdm_load_3d(xL,
                (unsigned long long)(const void*)
                    (x + ((((unsigned long long)b * C_IN + w * 2) * HW + rowBase) * HW)),
                /*td*/ HW, (unsigned)nrows, 2,
                /*tl*/ HW, (unsigned)nrows, 2,
                /*s0*/ HW, /*s1*/ HW * HW);
  }
  __builtin_amdgcn_s_wait_tensorcnt(0);   // TENSORcnt == 0 -> LDS tiles ready
  __syncthreads();
#else
  for (int i = tid; i < OGRP * KK; i += 256) thS[i] = theta[og * OGRP * KK + i];
  for (int i = tid; i < C_IN * nrows * HW; i += 256) {
    const int cc = i % HW, rest = i / HW;
    const int rr = rest % nrows, c = rest / nrows;
    xs[i] = x[((((unsigned long long)b * C_IN + c) * HW) + (rowBase + rr)) * HW + cc];
  }
  __syncthreads();
#endif

  // Pre-scale theta by SC = INV*log2(e) so the inner loop is a plain subtract.
  for (int i = tid; i < OGRP * KK; i += 256) thS[i] *= SC_F;
  __syncthreads();

  const int r    = r0 + (tid >> 5);       // output row (8 rows, one per wave)
  const int wcol = tid & 31;              // output col (32 lanes)

  float acc[OGRP];
#pragma unroll
  for (int o = 0; o < OGRP; ++o) acc[o] = 0.0f;

  for (int c = 0; c < C_IN; ++c) {
    // 3x3 patch (zero-padded at borders), pre-scaled by SC.
    float pi[9];
    const int cbase = c * nrows * HW;
#pragma unroll
    for (int i = 0; i < 3; ++i) {
      const int rr  = r + i - 1;
      const bool rok = (rr >= 0) && (rr < HW);
      const int rb  = cbase + (rr - rowBase) * HW;
#pragma unroll
      for (int j = 0; j < 3; ++j) {
        const int cc = wcol + j - 1;
        const bool ok = rok && (cc >= 0) && (cc < HW);
        pi[i * 3 + j] = (ok ? xs[rb + cc] : 0.0f) * SC_F;
      }
    }
    const float* th = &thS[c * 9];
#pragma unroll
    for (int o = 0; o < OGRP; ++o) {
      const float* tho = th + o * KK;     // wave-uniform LDS address -> broadcast
#pragma unroll
      for (int t = 0; t < 9; ++t) {
        // z = (p - theta) * INV * log2e; softplus(x) = ln2 * log2(1 + 2^z)
        const float z1 = pi[t] - tho[t];
        const float L1 = log2_raw(1.0f + exp2_raw(clamp2(z1)));
        const float L2 = log2_raw(1.0f + exp2_raw(clamp2(z1 - DLT2_F)));
        acc[o] = __builtin_fmaf(L1, L1, acc[o]);
        acc[o] = __builtin_fmaf(-L2, L2, acc[o]);
      }
    }
  }

  float* outp = out + (((unsigned long long)b * O_TOT + og * OGRP) * (HW * HW))
                    + r * HW + wcol;
#pragma unroll
  for (int o = 0; o < OGRP; ++o)
    outp[(unsigned long long)o * (HW * HW)] = OUTSC_F * acc[o];
}

extern "C" void kernel_launch(void* const* d_in, const int* in_sizes, int n_in,
                              void* d_out, int out_size, void* d_ws, size_t ws_size,
                              hipStream_t stream) {
  const float* x     = (const float*)d_in[0];   // 8*16*32*32
  const float* theta = (const float*)d_in[1];   // 64*144
  float* out = (float*)d_out;                   // 8*64*32*32
  (void)in_sizes; (void)n_in; (void)out_size; (void)d_ws; (void)ws_size;
  dim3 grid(B_N * 8 * 4);   // 256 blocks: batch x o-group x row-quarter
  dim3 block(256);          // 8 wave32 = 8 rows x 32 cols
  ekv_conv2d_kernel<<<grid, block, 0, stream>>>(x, theta, out);
}